// Memory_37426345017526
// MI455X (gfx1250) — compile-verified
//
#include <hip/hip_runtime.h>

// Problem constants (must match reference)
#define T_DIM   4096
#define N_DIM   4096
#define NBITS   16
#define WPN     2115        // ceil(65536 / 31)
#define ROWS    4           // batch rows per workgroup
#define TILE    256         // neurons per tile (== blockDim.x)
#define NTILES  (N_DIM / TILE)   // 16
#define TILE_BYTES (TILE * NBITS * 8)   // 32 KB per connections tile

// Native vector type (HIP's int4 is a class and is rejected by
// __builtin_nontemporal_load).
typedef int  v4i __attribute__((ext_vector_type(4)));
typedef unsigned long long u64_t;

// 32-bit LDS byte address from a generic __shared__ pointer.
// Flat pointers into the LDS aperture carry the LDS byte address in their
// low 32 bits.
__device__ __forceinline__ unsigned lds_addr_of(const void* p) {
    return (unsigned)(size_t)p;
}

// CDNA5 async global->LDS copy, 16 B per lane, tracked by ASYNCcnt.
// The instruction offset applies to BOTH the LDS and global addresses
// so an identity-mapped tile copy can
// share one address VGPR pair across all 8 chunks and vary only `offset:`.
template <int OFF>
__device__ __forceinline__ void async_g2l_b128(unsigned lds, const void* gaddr) {
    asm volatile("global_load_async_to_lds_b128 %0, %1, off offset:%c2"
                 :: "v"(lds), "v"(gaddr), "i"(OFF)
                 : "memory");
}

// Stage one 32 KB connections tile: 8 async b128 ops per lane, one shared
// address pair, offsets 0..0x7000.
__device__ __forceinline__ void stage_tile(unsigned lds_base, const void* g) {
    async_g2l_b128<0x0000>(lds_base, g);
    async_g2l_b128<0x1000>(lds_base, g);
    async_g2l_b128<0x2000>(lds_base, g);
    async_g2l_b128<0x3000>(lds_base, g);
    async_g2l_b128<0x4000>(lds_base, g);
    async_g2l_b128<0x5000>(lds_base, g);
    async_g2l_b128<0x6000>(lds_base, g);
    async_g2l_b128<0x7000>(lds_base, g);
}

__device__ __forceinline__ void wait_async_le8() {
    asm volatile("s_wait_asynccnt 8" ::: "memory");
}
__device__ __forceinline__ void wait_async_le0() {
    asm volatile("s_wait_asynccnt 0" ::: "memory");
}

__global__ __launch_bounds__(TILE) void wnn_memory_kernel(
    const int*   __restrict__ x,     // (B, T) int32, values 0/1
    const u64_t* __restrict__ conn,  // (N, 16) int64, values < T
    const u64_t* __restrict__ mem,   // (N, WPN) int64
    float*       __restrict__ out)   // (B, N)
{
    // 4 rows x 4096 bits = 2 KB of packed x bits
    __shared__ unsigned int s_bits[ROWS * (T_DIM / 32)];
    // double-buffered connections tile: 2 x 256 neurons x 16 x 8B = 64 KB
    __shared__ __align__(16) u64_t s_conn[2][TILE * NBITS];

    const int tid = threadIdx.x;
    const int b0  = blockIdx.x * ROWS;

    // ---- Stage tile 0 asynchronously ----
    stage_tile(lds_addr_of(&s_conn[0][0]) + tid * 16,
               (const char*)conn + (size_t)tid * 16);

    // ---- Pack x rows into LDS bit arrays (overlaps async staging).
    // x rows are touched by exactly one WG -> non-temporal loads keep
    // conn/memory_words resident in L2.
    #pragma unroll
    for (int r = 0; r < ROWS; ++r) {
        const v4i* xr = (const v4i*)(x + (size_t)(b0 + r) * T_DIM) + tid * 4;
        unsigned m = 0;
        #pragma unroll
        for (int j = 0; j < 4; ++j) {
            v4i v = __builtin_nontemporal_load(xr + j);
            m |= (unsigned)(v.x & 1) << (j * 4 + 0);
            m |= (unsigned)(v.y & 1) << (j * 4 + 1);
            m |= (unsigned)(v.z & 1) << (j * 4 + 2);
            m |= (unsigned)(v.w & 1) << (j * 4 + 3);
        }
        // element (tid*16 + e) of row r lands at bit (tid*16 + e) of the row
        ((unsigned short*)s_bits)[r * TILE + tid] = (unsigned short)m;
    }

    // ---- Main loop over neuron tiles, double-buffered ----
    for (int t = 0; t < NTILES; ++t) {
        const int buf = t & 1;

        if (t + 1 < NTILES) {
            // prefetch next tile into the other buffer
            stage_tile(lds_addr_of(&s_conn[buf ^ 1][0]) + tid * 16,
                       (const char*)conn + (size_t)(t + 1) * TILE_BYTES
                                         + (size_t)tid * 16);
            // 8 outstanding = next tile only; current tile's 8 are complete
            // (async loads complete in order within a wave)
            wait_async_le8();
        } else {
            wait_async_le0();
        }
        __syncthreads();   // tile t resident for all waves; bits packed

        // one neuron per thread in this tile, ROWS outputs per thread;
        // read the 16 indices as 4 x 32 B ulonglong4 loads
        const ulonglong4* cp4 =
            (const ulonglong4*)&s_conn[buf][(size_t)tid * NBITS];
        unsigned a[ROWS];
        #pragma unroll
        for (int r = 0; r < ROWS; ++r) a[r] = 0u;

        #pragma unroll
        for (int q = 0; q < 4; ++q) {
            ulonglong4 c = cp4[q];
            u64_t idx4[4] = {c.x, c.y, c.z, c.w};
            #pragma unroll
            for (int e = 0; e < 4; ++e) {
                unsigned idx = (unsigned)idx4[e];        // < 4096
                unsigned w = idx >> 5, s = idx & 31u;
                #pragma unroll
                for (int r = 0; r < ROWS; ++r)
                    a[r] = (a[r] << 1) |
                           ((s_bits[r * (T_DIM / 32) + w] >> s) & 1u);
            }
        }

        const int n = t * TILE + tid;
        #pragma unroll
        for (int r = 0; r < ROWS; ++r) {
            unsigned addr = a[r];                    // 16-bit address, MSB-first
            unsigned wi = addr / 31u;                // mul-hi by compiler
            unsigned sh = (addr - wi * 31u) * 2u;
            u64_t word = mem[(size_t)n * WPN + wi];  // L2-resident gather
            float v = (float)((word >> sh) & 3ull);
            // write-once output: non-temporal store, keep L2 for gathers
            __builtin_nontemporal_store(v, &out[(size_t)(b0 + r) * N_DIM + n]);
        }
        __syncthreads();   // everyone done reading buf before tile t+2 overwrites it
    }
}

extern "C" void kernel_launch(void* const* d_in, const int* in_sizes, int n_in,
                              void* d_out, int out_size, void* d_ws, size_t ws_size,
                              hipStream_t stream) {
    (void)in_sizes; (void)n_in; (void)out_size; (void)d_ws; (void)ws_size;
    const int*   x    = (const int*)d_in[0];
    const u64_t* conn = (const u64_t*)d_in[1];
    const u64_t* mem  = (const u64_t*)d_in[2];
    float*       out  = (float*)d_out;

    dim3 grid(1024 / ROWS);   // B / ROWS = 256 workgroups
    dim3 block(TILE);         // 256 threads = 8 wave32
    wnn_memory_kernel<<<grid, block, 0, stream>>>(x, conn, mem, out);
}